// RelativeDifferenceLoss_6940667150514
// MI455X (gfx1250) — compile-verified
//
#include <hip/hip_runtime.h>

typedef float f4 __attribute__((ext_vector_type(4)));

// rel = |x-y|          if y == 0
//       |x-y| / y      otherwise   (signed y, matching the reference)
__device__ __forceinline__ float rel4(f4 xv, f4 yv) {
    float s = 0.0f;
#pragma unroll
    for (int k = 0; k < 4; ++k) {
        float d  = fabsf(xv[k] - yv[k]);
        float rq = d * __builtin_amdgcn_rcpf(yv[k]);   // v_rcp_f32 (TRANS, co-executes)
        s += (yv[k] == 0.0f) ? d : rq;
    }
    return s;
}

__device__ __forceinline__ float wave32_reduce(float v) {
#pragma unroll
    for (int m = 16; m >= 1; m >>= 1)
        v += __shfl_xor(v, m, 32);       // wave32-only on gfx1250
    return v;
}

__global__ void __launch_bounds__(256)
reldiff_partial_kernel(const f4* __restrict__ x, const f4* __restrict__ y,
                       float* __restrict__ partial, int n4) {
    const int tid    = blockIdx.x * blockDim.x + threadIdx.x;
    const int stride = gridDim.x * blockDim.x;

    float acc = 0.0f;
    int i = tid;
    // 4-way unrolled: 8 independent global_load_b128 in flight per wave
    for (; i + 3 * stride < n4; i += 4 * stride) {
        __builtin_prefetch(&x[i + 8 * stride], 0, 0);   // global_prefetch_b8 (speculative)
        __builtin_prefetch(&y[i + 8 * stride], 0, 0);
        f4 x0 = __builtin_nontemporal_load(&x[i]);
        f4 x1 = __builtin_nontemporal_load(&x[i + stride]);
        f4 x2 = __builtin_nontemporal_load(&x[i + 2 * stride]);
        f4 x3 = __builtin_nontemporal_load(&x[i + 3 * stride]);
        f4 y0 = __builtin_nontemporal_load(&y[i]);
        f4 y1 = __builtin_nontemporal_load(&y[i + stride]);
        f4 y2 = __builtin_nontemporal_load(&y[i + 2 * stride]);
        f4 y3 = __builtin_nontemporal_load(&y[i + 3 * stride]);
        acc += rel4(x0, y0);
        acc += rel4(x1, y1);
        acc += rel4(x2, y2);
        acc += rel4(x3, y3);
    }
    for (; i < n4; i += stride) {   // remainder (empty for N = 2^25 with this grid)
        f4 xv = __builtin_nontemporal_load(&x[i]);
        f4 yv = __builtin_nontemporal_load(&y[i]);
        acc += rel4(xv, yv);
    }

    acc = wave32_reduce(acc);

    __shared__ float lds[8];                       // 256 threads = 8 wave32
    const int lane = threadIdx.x & 31;
    const int wave = threadIdx.x >> 5;
    if (lane == 0) lds[wave] = acc;
    __syncthreads();
    if (threadIdx.x == 0) {
        float s = 0.0f;
#pragma unroll
        for (int w = 0; w < 8; ++w) s += lds[w];
        partial[blockIdx.x] = s;                   // fully rewritten every call
    }
}

__global__ void __launch_bounds__(256)
reldiff_final_kernel(const float* __restrict__ partial, float* __restrict__ out,
                     int np, float invN) {
    float acc = 0.0f;
    for (int i = threadIdx.x; i < np; i += blockDim.x)
        acc += partial[i];

    acc = wave32_reduce(acc);

    __shared__ float lds[8];
    const int lane = threadIdx.x & 31;
    const int wave = threadIdx.x >> 5;
    if (lane == 0) lds[wave] = acc;
    __syncthreads();
    if (threadIdx.x == 0) {
        float s = 0.0f;
#pragma unroll
        for (int w = 0; w < 8; ++w) s += lds[w];
        out[0] = s * invN;                         // mean
    }
}

extern "C" void kernel_launch(void* const* d_in, const int* in_sizes, int n_in,
                              void* d_out, int out_size, void* d_ws, size_t ws_size,
                              hipStream_t stream) {
    const f4* x = (const f4*)d_in[0];
    const f4* y = (const f4*)d_in[1];
    const int n  = in_sizes[0];        // 2^25
    const int n4 = n >> 2;             // float4 count

    constexpr int kBlocks  = 2048;
    constexpr int kThreads = 256;
    float* partial = (float*)d_ws;     // 2048 floats = 8 KB scratch

    reldiff_partial_kernel<<<kBlocks, kThreads, 0, stream>>>(x, y, partial, n4);
    reldiff_final_kernel<<<1, kThreads, 0, stream>>>(partial, (float*)d_out,
                                                     kBlocks, 1.0f / (float)n);
}